// SoftGraphCut_57251914055802
// MI455X (gfx1250) — compile-verified
//
#include <hip/hip_runtime.h>

// ---------------------------------------------------------------------------
// Exact block-tridiagonal (block-Thomas) direct solve of A x = b on a 64x64
// grid, 4 batches x 21 RHS. Diagonal 64x64 blocks are tridiagonal, coupling
// blocks are diagonal, so the Schur recursion only needs:
//   * 64x64 Gauss-Jordan inversions (VALU, LDS-resident, no pivoting: SPD,
//     strictly diagonally dominant by 1)
//   * 64x64 @ 64x32 fp32 GEMMs  -> V_WMMA_F32_16X16X4_F32 tiles
// One workgroup (1024 threads = 32 wave32) per batch.
// Global<->LDS traffic for the T_i / y-hat spills uses the CDNA5 async
// LDS DMA path (GLOBAL_{LOAD,STORE}_ASYNC_*_LDS_B128, ASYNCcnt) when the
// toolchain exposes the builtins, with a plain-VGPR fallback otherwise.
// ---------------------------------------------------------------------------

typedef __attribute__((ext_vector_type(2))) float v2f;
typedef __attribute__((ext_vector_type(8))) float v8f;
typedef __attribute__((ext_vector_type(4))) int   v4i;

#define HH 64
#define WW 64
#define CC 21
#define NP 32          // padded RHS count (2 WMMA N-tiles); cols 21..31 stay 0
#define NTHREADS 1024

// ---- gfx1250 async global<->LDS support probes ---------------------------
#if defined(__has_builtin)
#if __has_builtin(__builtin_amdgcn_global_load_async_to_lds_b128)
#define HAVE_ASYNC_LOAD 1
#endif
#if __has_builtin(__builtin_amdgcn_global_store_async_from_lds_b128)
#define HAVE_ASYNC_STORE 1
#endif
#if __has_builtin(__builtin_amdgcn_s_wait_asynccnt)
#define HAVE_WAIT_ASYNC_BUILTIN 1
#endif
#endif

#ifndef HAVE_ASYNC_LOAD
#define HAVE_ASYNC_LOAD 0
#endif
#ifndef HAVE_ASYNC_STORE
#define HAVE_ASYNC_STORE 0
#endif

#if HAVE_ASYNC_LOAD || HAVE_ASYNC_STORE
// Builtin signature (from clang diagnostics): arg0 = int4 addrspace(1)*,
// arg1 = int4 addrspace(3)*, then imm offset, imm cpol.
using g_v4i = __attribute__((address_space(1))) v4i;
using l_v4i = __attribute__((address_space(3))) v4i;
// Generic->addrspace casts: AMDGPU generic LDS addresses carry the LDS byte
// offset in the low 32 bits (ISA 10.2 aperture calc), so integer truncation
// is the canonical conversion; global pointers are value-identical.
__device__ __forceinline__ g_v4i* to_as1(const void* p) {
  return (g_v4i*)(unsigned long long)p;
}
__device__ __forceinline__ l_v4i* to_as3(const void* p) {
  return (l_v4i*)(unsigned int)(unsigned long long)p;
}
__device__ __forceinline__ void wait_async0() {
#if defined(HAVE_WAIT_ASYNC_BUILTIN)
  __builtin_amdgcn_s_wait_asynccnt(0);
#else
  asm volatile("s_wait_asynccnt 0x0" ::: "memory");
#endif
}
#endif

// ---- 64x64 @ 64xNP fp32 GEMM, Out = Tm @ Z, via V_WMMA_F32_16X16X4_F32 ----
// Fragment layout assumptions (CDNA5 ISA 7.12.2, 32-bit operands, wave32):
//   A 16x4 : lanes 0-15 -> M=lane, K={0,1} in v0,v1 ; lanes 16-31 -> K={2,3}
//   B 4x16 : lanes 0-15 -> N=lane, K={0,1} in v0,v1 ; lanes 16-31 -> K={2,3}
//   C/D    : VGPR r -> M=r (lanes 0-15) / M=r+8 (lanes 16-31), N=lane%16
__device__ __forceinline__ void gemm64(const float* __restrict__ Tm,
                                       const float* __restrict__ Z,
                                       float* __restrict__ Out, int tid) {
  const int wave = tid >> 5;
  if (wave >= 8) return;          // wave-uniform branch: EXEC all-1s inside
  const int lane  = tid & 31;
  const int mt    = wave >> 1;    // 0..3  (M tile)
  const int nt    = wave & 1;     // 0..1  (N tile)
  const int l15   = lane & 15;
  const int khalf = (lane >> 4) << 1;   // 0 or 2

  v8f acc = {0.f, 0.f, 0.f, 0.f, 0.f, 0.f, 0.f, 0.f};
  const int arow = (mt * 16 + l15) * 64;
  const int bcol = nt * 16 + l15;
#pragma unroll
  for (int k0 = 0; k0 < 64; k0 += 4) {
    const int kk = k0 + khalf;
    v2f a, b;
    a.x = Tm[arow + kk];
    a.y = Tm[arow + kk + 1];
    b.x = Z[kk * NP + bcol];
    b.y = Z[(kk + 1) * NP + bcol];
    acc = __builtin_amdgcn_wmma_f32_16x16x4_f32(
        /*neg_a=*/false, a, /*neg_b=*/false, b,
        /*c_mod=*/(short)0, acc, /*reuse_a=*/false, /*reuse_b=*/false);
  }
  const int mbase = mt * 16 + ((lane >> 4) ? 8 : 0);
#pragma unroll
  for (int r = 0; r < 8; ++r) Out[(mbase + r) * NP + bcol] = acc[r];
}

// ---- In-place Gauss-Jordan inversion of 64x64 SPD matrix in LDS ----------
__device__ __forceinline__ void invert64(float* S, int tid) {
  for (int p = 0; p < 64; ++p) {
    __syncthreads();
    const float piv = 1.0f / S[p * 64 + p];
    // scale pivot row (except pivot element)
    if (tid < 64 && tid != p) S[p * 64 + tid] *= piv;
    __syncthreads();
    // rank-1 elimination of all other rows (reads col p & row p, writes rest)
    for (int idx = tid; idx < 64 * 64; idx += NTHREADS) {
      const int r = idx >> 6, j = idx & 63;
      if (r != p && j != p) S[idx] -= S[r * 64 + p] * S[p * 64 + j];
    }
    __syncthreads();
    // finish column p
    if (tid < 64) {
      if (tid != p) S[tid * 64 + p] = -S[tid * 64 + p] * piv;
      else          S[p * 64 + p]   = piv;
    }
  }
  __syncthreads();
}

__global__ __launch_bounds__(NTHREADS, 1) void softgraphcut_block_thomas(
    const float* __restrict__ ps,    // [4,21,64,64]
    const float* __restrict__ wcol,  // [4,63,64]
    const float* __restrict__ wrow,  // [4,64,63]
    float* __restrict__ out,         // [4,21,64,64]
    float* __restrict__ Tws,         // [4,64,64,64]  stored S_i^{-1}
    float* __restrict__ Yws) {       // [4,64,64,NP]  stored y-hat
  __shared__ alignas(16) float S[64 * 64];   // Schur complement / its inverse
  __shared__ alignas(16) float Tp[64 * 64];  // T_{i-1} (fwd) or T_i (bwd)
  __shared__ alignas(16) float Ya[64 * NP];  // y-hat_{i-1} (fwd) / Z (bwd)
  __shared__ alignas(16) float Yb[64 * NP];  // y-hat_i (fwd)    / x (bwd)
  __shared__ float wprev[64];
  __shared__ float wrowi[64];

  const int tid = threadIdx.x;
  const int b   = blockIdx.x;

  // init
  for (int idx = tid; idx < 64 * 64; idx += NTHREADS) Tp[idx] = 0.0f;
  for (int idx = tid; idx < 64 * NP; idx += NTHREADS) Ya[idx] = 0.0f;
  __syncthreads();

  // ---------------- forward sweep ----------------
  for (int i = 0; i < HH; ++i) {
    if (tid < 64) {
      wprev[tid] = (i > 0)   ? wcol[(b * 63 + (i - 1)) * 64 + tid] : 0.0f;
      wrowi[tid] = (tid < 63) ? wrow[(b * 64 + i) * 63 + tid]      : 0.0f;
    }
    __syncthreads();

    // raw coupling GEMM: Yb = T_{i-1} @ yhat_{i-1}   (zeros when i==0)
    gemm64(Tp, Ya, Yb, tid);
    __syncthreads();

    // yhat_i = b_i + diag(wprev) * (T_{i-1} @ yhat_{i-1}); keep cols>=21 at 0
    for (int idx = tid; idx < 64 * NP; idx += NTHREADS) {
      const int r = idx >> 5, ch = idx & 31;
      float v = 0.0f;
      if (ch < CC)
        v = ps[((b * CC + ch) * HH + i) * WW + r] + wprev[r] * Yb[idx];
      Yb[idx] = v;
    }
    // S_i = D_i - diag(wprev) T_{i-1} diag(wprev)
    for (int idx = tid; idx < 64 * 64; idx += NTHREADS) {
      const int r = idx >> 6, cc = idx & 63;
      float d = 0.0f;
      if (r == cc) {
        const float wca = (i < 63) ? wcol[(b * 63 + i) * 64 + r] : 0.0f;
        const float wrb = (r > 0) ? wrowi[r - 1] : 0.0f;
        d = 1.0f + wprev[r] + wca + wrb + wrowi[r];
      } else if (cc == r + 1) {
        d = -wrowi[r];
      } else if (cc + 1 == r) {
        d = -wrowi[cc];
      }
      S[idx] = d - wprev[r] * wprev[cc] * Tp[idx];
    }
    // (invert64 begins with __syncthreads)
    invert64(S, tid);                 // S := T_i

    // spill T_i (16KB) and yhat_i (8KB) to workspace; async DMA straight from
    // LDS when available (overlapped with the Tp/Ya staging copies below)
    float* Tdst = &Tws[((size_t)(b * HH + i)) << 12];
    float* Ydst = &Yws[(size_t)(b * HH + i) * (64 * NP)];
#if HAVE_ASYNC_STORE
    __builtin_amdgcn_global_store_async_from_lds_b128(
        to_as1(Tdst + tid * 4), to_as3(&S[tid * 4]), 0, 0);
    if (tid < (64 * NP) / 4)
      __builtin_amdgcn_global_store_async_from_lds_b128(
          to_as1(Ydst + tid * 4), to_as3(&Yb[tid * 4]), 0, 0);
#endif
    // stage T_i and yhat_i in LDS for the next step
    for (int idx = tid; idx < 64 * 64; idx += NTHREADS) {
      const float t = S[idx];
      Tp[idx] = t;
#if !HAVE_ASYNC_STORE
      Tdst[idx] = t;
#endif
    }
    for (int idx = tid; idx < 64 * NP; idx += NTHREADS) {
      const float y = Yb[idx];
      Ya[idx] = y;
#if !HAVE_ASYNC_STORE
      Ydst[idx] = y;
#endif
    }
#if HAVE_ASYNC_STORE
    wait_async0();   // S / Yb are rewritten next iteration
#endif
    __syncthreads();
  }

  // ---------------- backward sweep ----------------
  for (int idx = tid; idx < 64 * NP; idx += NTHREADS) Yb[idx] = 0.0f; // x_{H}
  __syncthreads();

  for (int i = HH - 1; i >= 0; --i) {
    if (tid < 64)
      wprev[tid] = (i < 63) ? wcol[(b * 63 + i) * 64 + tid] : 0.0f;

    const float* Tsrc = &Tws[((size_t)(b * HH + i)) << 12];
#if HAVE_ASYNC_LOAD
    // async DMA T_i into LDS: 1024 threads x 16B = 16KB, no VGPR staging
    __builtin_amdgcn_global_load_async_to_lds_b128(
        to_as1(Tsrc + tid * 4), to_as3(&Tp[tid * 4]), 0, 0);
#else
    for (int idx = tid; idx < 64 * 64; idx += NTHREADS) Tp[idx] = Tsrc[idx];
#endif
    // prefetch next iteration's T into L2/LDS-path caches behind this GEMM
    if (i > 0)
      __builtin_prefetch(&Tws[(((size_t)(b * HH + i - 1)) << 12) + tid * 4],
                         0, 3);
#if HAVE_ASYNC_LOAD
    wait_async0();
#endif
    __syncthreads();

    // Z = yhat_i + diag(w_i) * x_{i+1}
    for (int idx = tid; idx < 64 * NP; idx += NTHREADS) {
      const int r = idx >> 5;
      Ya[idx] = Yws[(size_t)(b * HH + i) * (64 * NP) + idx] + wprev[r] * Yb[idx];
    }
    __syncthreads();

    // x_i = T_i @ Z
    gemm64(Tp, Ya, Yb, tid);
    __syncthreads();

    // scatter x_i to output [b][ch][i][r]
    for (int idx = tid; idx < 64 * CC; idx += NTHREADS) {
      const int r = idx / CC, ch = idx % CC;
      out[((b * CC + ch) * HH + i) * WW + r] = Yb[r * NP + ch];
    }
    __syncthreads();
  }
}

extern "C" void kernel_launch(void* const* d_in, const int* in_sizes, int n_in,
                              void* d_out, int out_size, void* d_ws,
                              size_t ws_size, hipStream_t stream) {
  (void)in_sizes; (void)n_in; (void)out_size; (void)ws_size;
  const float* ps   = (const float*)d_in[0];  // [4,21,64,64]
  const float* wcol = (const float*)d_in[1];  // [4,63,64]
  const float* wrow = (const float*)d_in[2];  // [4,64,63]
  float* out = (float*)d_out;                 // [4,21,64,64]

  // workspace: T inverses (4 MB) then y-hat (2 MB)
  float* Tws = (float*)d_ws;
  float* Yws = Tws + (size_t)4 * HH * 64 * 64;

  softgraphcut_block_thomas<<<dim3(4), dim3(NTHREADS), 0, stream>>>(
      ps, wcol, wrow, out, Tws, Yws);
}